// Division_Tree_3169685864551
// MI455X (gfx1250) — compile-verified
//
#include <hip/hip_runtime.h>
#include <math.h>

typedef float v2f __attribute__((ext_vector_type(2)));
typedef float v8f __attribute__((ext_vector_type(8)));

#define N_ROWS       65536
#define FMAP         64
#define LEAF         512
#define NLEAF        128
#define NODES_TOTAL  65663
#define NODES_ELEMS  (NODES_TOTAL * 64)          // 4202432
#define NNZ          34473347
#define IDX_BASE     NODES_ELEMS                  // int32 region start (elements)
#define VAL_BASE     (NODES_ELEMS + 2 * NNZ)      // 73149126

// Subtree node counts T[d] and value counts VC[d] for depth d = 1..8.
__constant__ int c_T[9]  = {0, 65663, 32831, 16415, 8207, 4103, 2051, 1025, 512};
__constant__ int c_VC[9] = {0, 34473347, 17171011, 8552675, 4259923,
                            2121755, 1056775, 526337, 262144};

// ---------------------------------------------------------------- row norms
__global__ void k_rownorm(const float* __restrict__ X, float* __restrict__ q) {
    int row = blockIdx.x * blockDim.x + threadIdx.x;
    if (row >= N_ROWS) return;
    const float4* p = (const float4*)(X + (size_t)row * FMAP);
    float s = 0.0f;
#pragma unroll
    for (int c = 0; c < 16; ++c) {
        float4 v = p[c];
        s += v.x * v.x + v.y * v.y + v.z * v.z + v.w * v.w;
    }
    q[row] = s;
}

// ------------------------------------------------------- per-leaf row sums
__global__ void k_leafsum(const float* __restrict__ X, float* __restrict__ lsum) {
    int leaf = blockIdx.x;            // 128 blocks
    int f    = threadIdx.x;           // 64 threads
    const float* p = X + (size_t)leaf * LEAF * FMAP + f;
    float s = 0.0f;
    for (int r = 0; r < LEAF; ++r) s += p[r * FMAP];
    lsum[leaf * FMAP + f] = s;
}

// ------------------------------------- bottom-up tree of means (127 nodes)
__global__ void k_tree(const float* __restrict__ lsum, float* __restrict__ intNodes) {
    __shared__ float ts[128 * 64];    // totalsum per live subtree, per feature
    int f = threadIdx.x;              // 64 threads, one per feature
    for (int s = 0; s < 128; ++s) ts[s * 64 + f] = lsum[s * 64 + f];
    for (int d = 7; d >= 1; --d) {
        int   cnt   = 1 << (d - 1);
        float inv_m = 1.0f / (float)(2 * c_T[d + 1]);
        int   hb    = cnt - 1;        // heap base of level d
        for (int j = 0; j < cnt; ++j) {
            float S    = ts[(2 * j) * 64 + f] + ts[(2 * j + 1) * 64 + f];
            float mean = S * inv_m;
            intNodes[(hb + j) * 64 + f] = mean;
            ts[j * 64 + f] = S + mean;
        }
    }
}

// ------------------------------------------------ emit all_nodes (output 0)
__global__ void k_nodes(const float* __restrict__ X,
                        const float* __restrict__ intNodes,
                        float* __restrict__ out) {
    int gid = blockIdx.x * 256 + threadIdx.x;
    if (gid >= NODES_ELEMS) return;
    int node = gid >> 6, f = gid & 63;
    int idx = node, j = 0, rowbase = 0, d = 1;
    const float* src = nullptr;
    while (d < 8) {
        int Tn = c_T[d + 1];
        if (idx == 2 * Tn) {                        // this subtree's mean node
            src = intNodes + (size_t)((1 << (d - 1)) - 1 + j) * 64;
            break;
        }
        if (idx < Tn) { j = 2 * j; }
        else          { idx -= Tn; rowbase += (N_ROWS >> d); j = 2 * j + 1; }
        ++d;
    }
    if (!src) src = X + (size_t)(rowbase + idx) * 64;   // leaf -> raw X row
    out[gid] = src[f];
}

// -------------------------- leaf Gram matrices via fp32 WMMA + fused exp
__global__ void __launch_bounds__(256)
k_leaf(const float* __restrict__ X, const float* __restrict__ q,
       float* __restrict__ out) {
    int wave = threadIdx.x >> 5;
    int lane = threadIdx.x & 31;
    int tile = blockIdx.x * 8 + wave;       // 131072 tiles total
    int leaf = tile >> 10;
    int t    = tile & 1023;
    int tm   = t >> 5, tn = t & 31;
    int hi   = lane >> 4, lm = lane & 15;

    const float* Abase = X + (size_t)leaf * LEAF * FMAP;
    const float* rowA  = Abase + (size_t)(tm * 16 + lm) * FMAP;
    const float* rowB  = Abase + (size_t)(tn * 16 + lm) * FMAP;

    // 16x4 f32 A layout: lane(0-15)=M, VGPR0/1 = K pair, lanes 16-31 hold K+2.
    v2f a[16], b[16];
#pragma unroll
    for (int c = 0; c < 16; ++c) {
        a[c] = *(const v2f*)(rowA + 4 * c + 2 * hi);
        b[c] = *(const v2f*)(rowB + 4 * c + 2 * hi);
    }
    v8f acc = {};
#pragma unroll
    for (int c = 0; c < 16; ++c)
        acc = __builtin_amdgcn_wmma_f32_16x16x4_f32(false, a[c], false, b[c],
                                                    (short)0, acc, false, false);

    // leaf node/value base offsets from the 7 path bits
    int NB = 0, VOff = 0;
#pragma unroll
    for (int k = 1; k <= 7; ++k) {
        int bit = (leaf >> (7 - k)) & 1;
        NB   += bit * c_T[k + 1];
        VOff += bit * c_VC[k + 1];
    }

    int   nl = tn * 16 + lm;
    float qn = q[leaf * LEAF + nl];
    float* val = out + VAL_BASE;
    int*   idx = (int*)out + IDX_BASE;
#pragma unroll
    for (int r = 0; r < 8; ++r) {
        int   ml = tm * 16 + r + 8 * hi;     // C layout: VGPR r -> M = r (+8 hi half)
        float qm = q[leaf * LEAF + ml];
        float v  = expf(-(qm + qn - 2.0f * acc[r]) * (1.0f / 64.0f));
        int   e  = VOff + ml * LEAF + nl;
        val[e]       = v;
        idx[e]       = NB + ml;
        idx[NNZ + e] = NB + nl;
    }
}

// ----------------------- internal-merge values/indices (node vs mean RBF)
__global__ void k_merge(const float* __restrict__ nodes, float* __restrict__ out) {
    int mid = blockIdx.y;                             // 0..126, heap order
    int d   = (31 - __builtin_clz(mid + 1)) + 1;      // depth of this merge
    int j   = (mid + 1) - (1 << (d - 1));             // index within level
    int m   = 2 * c_T[d + 1];                         // nodes before new mean
    if (blockIdx.x * 256 > m) return;                 // uniform whole-block exit

    int NB = 0, VOff = 0;
    for (int k = 1; k <= d - 1; ++k) {
        int bit = (j >> (d - 1 - k)) & 1;
        NB   += bit * c_T[k + 1];
        VOff += bit * c_VC[k + 1];
    }
    int vbase = VOff + 2 * c_VC[d + 1];

    __shared__ float mean[64];
    if (threadIdx.x < 16)
        ((float4*)mean)[threadIdx.x] =
            ((const float4*)(nodes + (size_t)(NB + m) * 64))[threadIdx.x];
    __syncthreads();

    int i = blockIdx.x * 256 + threadIdx.x;
    if (i > m) return;

    float* val = out + VAL_BASE;
    int*   idx = (int*)out + IDX_BASE;
    if (i == m) {                                     // trailing (m,m,1.0) entry
        val[vbase + 2 * m]       = 1.0f;
        idx[vbase + 2 * m]       = NB + m;
        idx[NNZ + vbase + 2 * m] = NB + m;
        return;
    }
    const float4* p = (const float4*)(nodes + (size_t)(NB + i) * 64);
    float d2 = 0.0f;
#pragma unroll
    for (int c = 0; c < 16; ++c) {
        float4 v  = p[c];
        float4 mu = ((const float4*)mean)[c];
        float dx = v.x - mu.x, dy = v.y - mu.y, dz = v.z - mu.z, dw = v.w - mu.w;
        d2 += dx * dx + dy * dy + dz * dz + dw * dw;
    }
    float v = expf(-d2 * (1.0f / 64.0f));
    val[vbase + i]           = v;
    idx[vbase + i]           = NB + i;
    idx[NNZ + vbase + i]     = NB + m;
    val[vbase + m + i]       = v;
    idx[vbase + m + i]       = NB + m;
    idx[NNZ + vbase + m + i] = NB + i;
}

extern "C" void kernel_launch(void* const* d_in, const int* in_sizes, int n_in,
                              void* d_out, int out_size, void* d_ws, size_t ws_size,
                              hipStream_t stream) {
    (void)in_sizes; (void)n_in; (void)out_size; (void)ws_size;
    const float* X   = (const float*)d_in[0];
    float*       out = (float*)d_out;
    float*       ws  = (float*)d_ws;
    float* q        = ws;                      // 65536 floats
    float* lsum     = ws + 65536;              // 128*64
    float* intNodes = ws + 65536 + 128 * 64;   // 127*64

    k_rownorm<<<N_ROWS / 256, 256, 0, stream>>>(X, q);
    k_leafsum<<<NLEAF, 64, 0, stream>>>(X, lsum);
    k_tree<<<1, 64, 0, stream>>>(lsum, intNodes);
    k_nodes<<<(NODES_ELEMS + 255) / 256, 256, 0, stream>>>(X, intNodes, out);
    k_leaf<<<131072 / 8, 256, 0, stream>>>(X, q, out);
    dim3 mg((65662 + 1 + 255) / 256, 127);
    k_merge<<<mg, 256, 0, stream>>>(out, out);
}